// D2STGNN_54597624266877
// MI455X (gfx1250) — compile-verified
//
#include <hip/hip_runtime.h>

// ---------------- model dims ----------------
#define B_     8
#define L_     12
#define N_     512
#define D_     32
#define ND_    10
#define TDQ_   10
#define KT_    3
#define FK_    256
#define OUTH_  512
#define NL_    5
#define HEADS_ 4
#define STEPS_ 4
#define GAP_   3
#define AF_    62
#define BN_    (B_*N_)
#define BNN_   ((long)B_*N_*N_)

typedef float v2f __attribute__((ext_vector_type(2)));
typedef float v8f __attribute__((ext_vector_type(8)));

static inline long cdiv(long a, long b) { return (a + b - 1) / b; }

// ================= WMMA batched GEMM =================
// C[bt] = act( alpha * A[bt/divA] @ (B[bt] or B[bt]^T) + bias )
// fp32 WMMA 16x16x4. Fragment layout (ISA 7.12.2):
//   A 16x4 : lane<16 -> K = k0,k0+1 ; lane>=16 -> K = k0+2,k0+3 (v2f, contiguous)
//   B  4x16: mirrored in K, N = lane&15
//   C/D    : 8 VGPRs, M = r + 8*(lane>>4), N = lane&15
// Full tiles take an unguarded, 4x-unrolled loop (4 wmma / iter, b64 loads);
// edge tiles and K%4 remainders fall back to the guarded loop.
template<int TRB, int ACT>
__global__ __launch_bounds__(128) void k_gemm(
    const float* __restrict__ A, const float* __restrict__ Bm,
    const float* __restrict__ bias, float* __restrict__ C,
    int M, int N, int K, int lda, int ldb, int ldc,
    long sA, long sB, long sC, int divA, float alpha)
{
  const int lane = threadIdx.x;          // 0..31
  const int wv   = threadIdx.y;          // 0..3 -> N sub-tile
  const int tm   = blockIdx.x * 16;
  const int tn   = (blockIdx.y * 4 + wv) * 16;
  const int bt   = blockIdx.z;
  const float* Ab = A  + (long)(bt / divA) * sA;
  const float* Bb = Bm + (long)bt * sB;
  float*       Cb = C  + (long)bt * sC;
  const int half = lane >> 4;
  const int l16  = lane & 15;
  const int am = tm + l16;
  const int bn = tn + l16;
  const bool full = (tm + 16 <= M) && (tn + 16 <= N);

  v8f acc = {0.f,0.f,0.f,0.f,0.f,0.f,0.f,0.f};
  int k0 = 0;

  if (full) {
    const float* ap = Ab + (long)am * lda + 2*half;      // this lane's A K-pair base
    const float* bp = TRB ? (Bb + (long)bn * ldb + 2*half) : (Bb + bn);
    // steady state: 16 K per iteration, 4 wmma back-to-back
    for (; k0 + 16 <= K; k0 += 16) {
      __builtin_prefetch(ap + k0 + 128, 0, 0);
#pragma unroll
      for (int u = 0; u < 4; ++u) {
        v2f av = *(const v2f*)(ap + k0 + 4*u);
        v2f bv;
        if (TRB) {
          bv = *(const v2f*)(bp + k0 + 4*u);
        } else {
          const int kk = k0 + 4*u + 2*half;
          bv.x = bp[(long)kk * ldb];
          bv.y = bp[(long)(kk+1) * ldb];
        }
        acc = __builtin_amdgcn_wmma_f32_16x16x4_f32(false, av, false, bv,
                                                    (short)0, acc, false, false);
      }
    }
    for (; k0 + 4 <= K; k0 += 4) {
      v2f av = *(const v2f*)(ap + k0);
      v2f bv;
      if (TRB) {
        bv = *(const v2f*)(bp + k0);
      } else {
        const int kk = k0 + 2*half;
        bv.x = bp[(long)kk * ldb];
        bv.y = bp[(long)(kk+1) * ldb];
      }
      acc = __builtin_amdgcn_wmma_f32_16x16x4_f32(false, av, false, bv,
                                                  (short)0, acc, false, false);
    }
  }

  // guarded tail: K remainder for full tiles, whole K for edge tiles
  for (; k0 < K; k0 += 4) {
    const int ka = k0 + 2*half;
    v2f av; av.x = 0.f; av.y = 0.f;
    if (am < M) {
      const float* ap = Ab + (long)am * lda + ka;
      if (ka + 1 < K)      { av.x = ap[0]; av.y = ap[1]; }
      else if (ka < K)     { av.x = ap[0]; }
    }
    v2f bv; bv.x = 0.f; bv.y = 0.f;
    if (bn < N) {
      if (TRB) {
        const float* bp = Bb + (long)bn * ldb + ka;
        if (ka + 1 < K)    { bv.x = bp[0]; bv.y = bp[1]; }
        else if (ka < K)   { bv.x = bp[0]; }
      } else {
        if (ka < K)        bv.x = Bb[(long)ka * ldb + bn];
        if (ka + 1 < K)    bv.y = Bb[(long)(ka+1) * ldb + bn];
      }
    }
    acc = __builtin_amdgcn_wmma_f32_16x16x4_f32(false, av, false, bv,
                                                (short)0, acc, false, false);
  }

  const int cn = tn + l16;
  const float bbias = (bias && cn < N) ? bias[cn] : 0.f;
  if (full) {
    float* cp = Cb + (long)(tm + 8*half) * ldc + cn;
#pragma unroll
    for (int r = 0; r < 8; ++r) {
      float v = acc[r] * alpha + bbias;
      if (ACT == 1) v = fmaxf(v, 0.f);
      else if (ACT == 2) v = 1.f / (1.f + __expf(-v));
      cp[(long)r * ldc] = v;
    }
  } else {
#pragma unroll
    for (int r = 0; r < 8; ++r) {
      const int cm = tm + r + 8*half;
      if (cm < M && cn < N) {
        float v = acc[r] * alpha + bbias;
        if (ACT == 1) v = fmaxf(v, 0.f);
        else if (ACT == 2) v = 1.f / (1.f + __expf(-v));
        Cb[(long)cm * ldc + cn] = v;
      }
    }
  }
}

static void gemm(hipStream_t st, const float* A, const float* Bm, const float* bias,
                 float* C, int M, int N, int K, int lda, int ldb, int ldc,
                 long sA, long sB, long sC, int batch, int divA, int transB,
                 int act, float alpha)
{
  dim3 blk(32, 4, 1);
  dim3 grd((unsigned)cdiv(M,16), (unsigned)cdiv(cdiv(N,16),4), (unsigned)batch);
#define KG_ARGS A, Bm, bias, C, M, N, K, lda, ldb, ldc, sA, sB, sC, divA, alpha
  if (transB) {
    if (act == 1)      k_gemm<1,1><<<grd, blk, 0, st>>>(KG_ARGS);
    else               k_gemm<1,0><<<grd, blk, 0, st>>>(KG_ARGS);
  } else {
    if (act == 1)      k_gemm<0,1><<<grd, blk, 0, st>>>(KG_ARGS);
    else if (act == 2) k_gemm<0,2><<<grd, blk, 0, st>>>(KG_ARGS);
    else               k_gemm<0,0><<<grd, blk, 0, st>>>(KG_ARGS);
  }
#undef KG_ARGS
}

// ================= elementwise / reduction kernels =================
#define GID ((long)blockIdx.x * blockDim.x + threadIdx.x)

__global__ void k_fill(float* p, long n, float v) { long i = GID; if (i < n) p[i] = v; }
__global__ void k_copy(float* d, const float* s, long n) { long i = GID; if (i < n) d[i] = s[i]; }
__global__ void k_add(float* d, const float* s, long n) { long i = GID; if (i < n) d[i] += s[i]; }
__global__ void k_relu_copy(float* d, const float* s, long n) { long i = GID; if (i < n) d[i] = fmaxf(s[i], 0.f); }

__global__ void k_time_embed(const float* __restrict__ hd, const float* __restrict__ TiD,
                             const float* __restrict__ DiW, float* TDb, float* DWb, float* hist)
{
  long i = GID; long tot = (long)B_*L_*N_; if (i >= tot) return;
  hist[i] = hd[i*3];
  int t = (int)(hd[i*3+1] * 288.f); t = t < 0 ? 0 : (t > 287 ? 287 : t);
  int w = (int)(hd[i*3+2]);         w = w < 0 ? 0 : (w > 6 ? 6 : w);
  for (int j = 0; j < TDQ_; ++j) { TDb[i*TDQ_+j] = TiD[t*TDQ_+j]; DWb[i*TDQ_+j] = DiW[w*TDQ_+j]; }
}

__global__ void k_pe(float* pe)
{
  long i = GID; if (i >= 64*D_) return;
  int pos = (int)(i / D_), d = (int)(i % D_);
  int dp = d & ~1;
  float div = __expf(-logf(10000.f) * (float)dp / (float)D_);
  float a = (float)pos * div;
  pe[i] = (d & 1) ? cosf(a) : sinf(a);
}

__global__ void k_xs2(const float* hist, float* Xs2)
{
  long i = GID; if (i >= (long)BN_*L_) return;
  int t = (int)(i % L_); long bn = i / L_;
  int b = (int)(bn / N_), n = (int)(bn % N_);
  Xs2[i] = hist[((long)b*L_ + t)*N_ + n];
}

__global__ void k_batchnorm(float* x, const float* g, const float* b, int rows, int cols)
{
  int c = blockIdx.x, t = threadIdx.x;
  __shared__ float s1[256], s2[256];
  float sum = 0.f, sq = 0.f;
  for (int i = t; i < rows; i += 256) { float v = x[(long)i*cols + c]; sum += v; sq += v*v; }
  s1[t] = sum; s2[t] = sq; __syncthreads();
  for (int o = 128; o > 0; o >>= 1) { if (t < o) { s1[t] += s1[t+o]; s2[t] += s2[t+o]; } __syncthreads(); }
  float mu = s1[0] / rows, var = s2[0] / rows - mu*mu;
  float inv = rsqrtf(var + 1e-5f), gg = g[c], bb = b[c];
  for (int i = t; i < rows; i += 256) { long k = (long)i*cols + c; x[k] = (x[k]-mu)*inv*gg + bb; }
}

__global__ void k_softmax(float* p, int cols)
{
  long row = blockIdx.x; float* r = p + row*cols; int t = threadIdx.x;
  __shared__ float sm[256];
  float m = -3.4e38f;
  for (int i = t; i < cols; i += 256) m = fmaxf(m, r[i]);
  sm[t] = m; __syncthreads();
  for (int o = 128; o > 0; o >>= 1) { if (t < o) sm[t] = fmaxf(sm[t], sm[t+o]); __syncthreads(); }
  m = sm[0]; __syncthreads();
  float s = 0.f;
  for (int i = t; i < cols; i += 256) { float e = __expf(r[i]-m); r[i] = e; s += e; }
  sm[t] = s; __syncthreads();
  for (int o = 128; o > 0; o >>= 1) { if (t < o) sm[t] += sm[t+o]; __syncthreads(); }
  float inv = 1.f / sm[0];
  for (int i = t; i < cols; i += 256) r[i] *= inv;
}

__global__ void k_build_xi(const float* dy, const float* TDb, const float* DWb,
                           const float* e, float* Xi)
{
  long i = GID; if (i >= BN_) return;
  int b = (int)(i / N_), n = (int)(i % N_);
  float* o = Xi + i*AF_;
  for (int j = 0; j < D_; ++j)   o[j]        = dy[i*D_+j];
  for (int j = 0; j < TDQ_; ++j) o[D_+j]     = TDb[(((long)b*L_+(L_-1))*N_+n)*TDQ_+j];
  for (int j = 0; j < TDQ_; ++j) o[D_+10+j]  = DWb[(((long)b*L_+(L_-1))*N_+n)*TDQ_+j];
  for (int j = 0; j < ND_; ++j)  o[D_+20+j]  = e[(long)n*ND_+j];
}

__global__ void k_graph_norm(const float* A, const float* adj, float* Gf, float* Gm)
{
  long row = blockIdx.x;           // b*N + n
  int n = (int)(row % N_);
  const float* a  = A   + row*N_;
  const float* ad = adj + (long)n*N_;
  float* gf = Gf + row*N_; float* gm = Gm + row*N_;
  int t = threadIdx.x;
  __shared__ float sm[256];
  float s = 0.f;
  for (int m = t; m < N_; m += 256) s += (ad[m] + 1e-7f) * a[m];
  sm[t] = s; __syncthreads();
  for (int o = 128; o > 0; o >>= 1) { if (t < o) sm[t] += sm[t+o]; __syncthreads(); }
  float inv = 1.f / sm[0];
  if (!(fabsf(inv) <= 3.4e38f)) inv = 0.f;
  for (int m = t; m < N_; m += 256) {
    float v = (ad[m] + 1e-7f) * a[m] * inv;
    gf[m] = v; gm[m] = (m == n) ? 0.f : v;
  }
}

__global__ void k_mask(const float* src, float* dst, long batch)
{
  long i = GID; long tot = batch*(long)N_*N_; if (i >= tot) return;
  long r = i / N_; int m = (int)(i % N_); int n = (int)(r % N_);
  dst[i] = (m == n) ? 0.f : src[i];
}

__global__ void k_embed_cur(const float* hist, const float* w, const float* b, float* cur)
{
  long i = GID; long tot = (long)B_*L_*N_*D_; if (i >= tot) return;
  int d = (int)(i % D_); long r = i / D_;
  cur[i] = hist[r]*w[d] + b[d];
}

__global__ void k_gate_feat(const float* TDb, const float* DWb, const float* Eu,
                            const float* Ed, float* f)
{
  long i = GID; long tot = (long)B_*L_*N_; if (i >= tot) return;
  int n = (int)(i % N_);
  float* o = f + i*40;
  for (int j = 0; j < TDQ_; ++j) o[j]    = TDb[i*TDQ_+j];
  for (int j = 0; j < TDQ_; ++j) o[10+j] = DWb[i*TDQ_+j];
  for (int j = 0; j < ND_; ++j)  o[20+j] = Eu[(long)n*ND_+j];
  for (int j = 0; j < ND_; ++j)  o[30+j] = Ed[(long)n*ND_+j];
}

__global__ void k_gate_apply(const float* cur, int Lc, const float* gateO, float* out)
{
  long i = GID; long tot = (long)B_*Lc*N_*D_; if (i >= tot) return;
  long r = i / D_; int n = (int)(r % N_); r /= N_;
  int l = (int)(r % Lc); int b = (int)(r / Lc);
  out[i] = cur[i] * gateO[((long)b*L_ + (L_-Lc) + l)*N_ + n];
}

__global__ void k_window(const float* X, int Lin, float* Xw, int L2)
{
  long i = GID; long tot = (long)B_*L2*N_*(KT_*D_); if (i >= tot) return;
  int c = (int)(i % (KT_*D_)); long r = i / (KT_*D_);
  int n = (int)(r % N_); r /= N_;
  int l = (int)(r % L2); int b = (int)(r / L2);
  int kt = c / D_, d = c % D_;
  Xw[i] = X[(((long)b*Lin + l + kt)*N_ + n)*D_ + d];
}

__global__ void k_mean_xsum(const float* o96, float* hs, float* Xsum, long rows)
{
  long i = GID; long tot = rows*D_; if (i >= tot) return;
  int d = (int)(i % D_); long r = i / D_;
  const float* p = o96 + r*(KT_*D_);
  float v = p[d] + p[D_+d] + p[2*D_+d];
  Xsum[i] = v;
  hs[r*(7*D_) + d] = v * (1.f/3.f);
}

__global__ void k_slice(float* dst, int dstL, int dstT, const float* src, int srcL,
                        int srcT, int cnt)
{
  long i = GID; long tot = (long)B_*cnt*N_*D_; if (i >= tot) return;
  int d = (int)(i % D_); long r = i / D_;
  int n = (int)(r % N_); r /= N_;
  int t = (int)(r % cnt); int b = (int)(r / cnt);
  dst[(((long)b*dstL + dstT + t)*N_ + n)*D_ + d] =
  src[(((long)b*srcL + srcT + t)*N_ + n)*D_ + d];
}

__global__ void k_btnd_lbnd(const float* X, float* out, int Ll)
{
  long i = GID; long tot = (long)Ll*BN_*D_; if (i >= tot) return;
  int t = (int)(i / ((long)BN_*D_)); long rem = i % ((long)BN_*D_);
  long bn = rem / D_; int d = (int)(rem % D_);
  int b = (int)(bn / N_), n = (int)(bn % N_);
  out[i] = X[(((long)b*Ll + t)*N_ + n)*D_ + d];
}

__global__ void k_add_pe(float* H, const float* pe, int len)
{
  long i = GID; long tot = (long)len*BN_*D_; if (i >= tot) return;
  int t = (int)(i / ((long)BN_*D_)); int d = (int)(i % D_);
  H[i] += pe[t*D_ + d];
}

__global__ void k_gru(const float* gi, const float* gh, const float* hp, float* ho)
{
  long i = GID; long tot = (long)BN_*D_; if (i >= tot) return;
  int d = (int)(i % D_); long bn = i / D_;
  const float* a = gi + bn*(3*D_); const float* h = gh + bn*(3*D_);
  float r = 1.f/(1.f+__expf(-(a[d]        + h[d])));
  float z = 1.f/(1.f+__expf(-(a[D_+d]     + h[D_+d])));
  float n = tanhf(a[2*D_+d] + r*h[2*D_+d]);
  ho[i] = (1.f - z)*n + z*hp[i];
}

__global__ void k_attn(const float* qp, const float* kp, const float* vp, float* o,
                       int Lq, int Lk)
{
  long idx = GID; long tot = (long)BN_*HEADS_*Lq; if (idx >= tot) return;
  int lq = (int)(idx % Lq); long r = idx / Lq;
  int h = (int)(r % HEADS_); long bn = r / HEADS_;
  float q[8];
  for (int d = 0; d < 8; ++d) q[d] = qp[((long)lq*BN_ + bn)*D_ + h*8 + d];
  float sc[16]; float mx = -3.4e38f;
  for (int lk = 0; lk < Lk; ++lk) {
    float s = 0.f;
    for (int d = 0; d < 8; ++d) s += q[d]*kp[((long)lk*BN_ + bn)*D_ + h*8 + d];
    s *= 0.3535533906f;
    sc[lk] = s; mx = fmaxf(mx, s);
  }
  float sum = 0.f;
  for (int lk = 0; lk < Lk; ++lk) { float e = __expf(sc[lk]-mx); sc[lk] = e; sum += e; }
  float inv = 1.f/sum;
  float acc[8] = {0,0,0,0,0,0,0,0};
  for (int lk = 0; lk < Lk; ++lk) {
    float w = sc[lk]*inv;
    for (int d = 0; d < 8; ++d) acc[d] += w*vp[((long)lk*BN_ + bn)*D_ + h*8 + d];
  }
  for (int d = 0; d < 8; ++d) o[((long)lq*BN_ + bn)*D_ + h*8 + d] = acc[d];
}

// row = (b,l,n), one wave per row (D_ == wave32), LN(cur[:, t0+l] - relu(back))
__global__ __launch_bounds__(256) void k_sub_relu_ln(
    const float* cur, int Lcur, int t0, const float* back, int backLayout,
    int L2, const float* g, const float* b, float* out)
{
  int lane = threadIdx.x;
  long row = (long)blockIdx.x * blockDim.y + threadIdx.y;
  long rows = (long)B_*L2*N_; if (row >= rows) return;
  int n = (int)(row % N_); long r2 = row / N_;
  int l = (int)(r2 % L2); int bb = (int)(r2 / L2);
  float cv = cur[(((long)bb*Lcur + t0 + l)*N_ + n)*D_ + lane];
  long bi = (backLayout == 0)
      ? ((((long)bb*L2 + l)*N_ + n)*D_ + lane)
      : (((long)l*BN_ + (long)bb*N_ + n)*D_ + lane);
  float v = cv - fmaxf(back[bi], 0.f);
  float s = v;
  for (int o = 16; o > 0; o >>= 1) s += __shfl_xor(s, o, 32);
  float mu = s * (1.f/32.f);
  float dv = v - mu; float q = dv*dv;
  for (int o = 16; o > 0; o >>= 1) q += __shfl_xor(q, o, 32);
  float var = q * (1.f/32.f);
  out[row*D_ + lane] = dv*rsqrtf(var + 1e-5f)*g[lane] + b[lane];
}

__global__ void k_fk_perm(float* fk_sum, const float* fkI)
{
  long i = GID; long tot = (long)STEPS_*BN_*FK_; if (i >= tot) return;
  int f = (int)(i % FK_); long row = i / FK_;
  int n = (int)(row % N_); row /= N_;
  int b = (int)(row % B_); int s = (int)(row / B_);
  fk_sum[((((long)b*STEPS_) + s)*N_ + n)*FK_ + f] += fkI[i];
}

__global__ void k_final(const float* o2, float* out)
{
  long i = GID; long tot = (long)B_*STEPS_*N_*GAP_; if (i >= tot) return;
  int j = (int)(i % GAP_); long r = i / GAP_;
  int n = (int)(r % N_); r /= N_;
  int s = (int)(r % STEPS_); int b = (int)(r / STEPS_);
  out[((long)b*N_ + n)*(STEPS_*GAP_) + s*GAP_ + j] = o2[i];
}

// ---------------- host helpers ----------------
#define EW(kern, total, ...) kern<<<dim3((unsigned)cdiv((total),256)), dim3(256), 0, stream>>>(__VA_ARGS__)

// st_conv: X(B,Lin,N,D) -> Cout batched (batch=B*L2, M=N, 32 cols), strideC given.
static void st_conv(hipStream_t stream, const float* Xin, int Lin,
                    const float* gm, const float* stam0, const float* stam1,
                    const float* fcW, const float* gcnW, const float* gcnb,
                    float* Xw, float* out96, float* Xsum, float* hs,
                    float* Cout, long strideC)
{
  int L2 = Lin - KT_ + 1;
  long rows = (long)B_*L2*N_;
  EW(k_window, rows*(KT_*D_), Xin, Lin, Xw, L2);
  gemm(stream, Xw, fcW, nullptr, out96, (int)rows, KT_*D_, KT_*D_,
       KT_*D_, KT_*D_, KT_*D_, 0, 0, 0, 1, 1, 0, 1, 1.f);
  EW(k_mean_xsum, rows*D_, out96, hs, Xsum, rows);
  int batch = B_*L2;
  for (int j = 0; j < 4; ++j)
    gemm(stream, gm + (long)j*BNN_, Xsum, nullptr, hs + D_*(1+j),
         N_, D_, N_, N_, D_, 7*D_, (long)N_*N_, (long)N_*D_, (long)N_*(7*D_),
         batch, L2, 0, 0, 1.f);
  gemm(stream, stam0, Xsum, nullptr, hs + D_*5, N_, D_, N_,
       N_, D_, 7*D_, 0, (long)N_*D_, (long)N_*(7*D_), batch, 1, 0, 0, 1.f);
  gemm(stream, stam1, Xsum, nullptr, hs + D_*6, N_, D_, N_,
       N_, D_, 7*D_, 0, (long)N_*D_, (long)N_*(7*D_), batch, 1, 0, 0, 1.f);
  gemm(stream, hs, gcnW, gcnb, Cout, N_, D_, 7*D_,
       7*D_, D_, D_, (long)N_*(7*D_), 0, strideC, batch, 1, 0, 0, 1.f);
}

extern "C" void kernel_launch(void* const* d_in, const int* in_sizes, int n_in,
                              void* d_out, int out_size, void* d_ws, size_t ws_size,
                              hipStream_t stream)
{
  (void)in_sizes; (void)n_in; (void)out_size; (void)ws_size;
  auto P = [&](int i) { return (const float*)d_in[i]; };
  // jax pytree (sorted-key) flatten order: history_data, params..., adjs
  const float* hd   = P(0);
  const float* DiW  = P(1);  const float* Ed = P(2); const float* Eu = P(3);
  const float* TiD  = P(4);
  const float* dWK  = P(5);  const float* dWQ = P(6);
  const float* bn_b = P(7);  const float* bn_g = P(8);
  const float* fc1W = P(9);  const float* fc1b = P(10);
  const float* fc2W = P(11); const float* fc2b = P(12);
  const float* embW = P(13); const float* embB = P(14);
  const float* out1W = P(150); const float* out1b = P(151);
  const float* out2W = P(152); const float* out2b = P(153);
  const float* adjs  = P(154);

  float* w = (float*)d_ws;
  size_t off = 0;
  auto WS = [&](long n) { float* p = w + off; off += (size_t)n; return p; };

  float* TDb   = WS((long)B_*L_*N_*TDQ_);
  float* DWb   = WS((long)B_*L_*N_*TDQ_);
  float* hist  = WS((long)B_*L_*N_);
  float* pe    = WS(64*D_);
  float* Xs2   = WS((long)BN_*L_);
  float* h64   = WS((long)BN_*64);
  float* dy    = WS((long)BN_*D_);
  float* Xi    = WS((long)BN_*AF_);
  float* Qm    = WS((long)BN_*D_);
  float* Km    = WS((long)BN_*D_);
  float* Alog  = WS(BNN_);
  float* gfull = WS(BNN_);
  float* gpow  = WS(BNN_);
  float* gm    = WS(4*BNN_);
  float* sta0  = WS((long)N_*N_);
  float* sta2  = WS((long)N_*N_);
  float* stam0 = WS((long)N_*N_);
  float* stam1 = WS((long)N_*N_);
  float* cur   = WS((long)B_*L_*N_*D_);
  float* gFeat = WS((long)B_*L_*N_*40);
  float* gH    = WS((long)B_*L_*N_*64);
  float* gO    = WS((long)B_*L_*N_);
  float* gated = WS((long)B_*L_*N_*D_);
  float* Xw    = WS((long)B_*10*N_*(KT_*D_));
  float* o96   = WS((long)B_*10*N_*(KT_*D_));
  float* Xsum  = WS((long)B_*10*N_*D_);
  float* hs    = WS((long)B_*10*N_*(7*D_));
  float* hdif  = WS((long)B_*10*N_*D_);
  float* predsD= WS((long)B_*STEPS_*N_*D_);
  float* Xcat  = WS((long)B_*KT_*N_*D_);
  float* fk_sum= WS((long)B_*STEPS_*N_*FK_);
  float* fk_tmp= WS((long)B_*STEPS_*N_*FK_);
  float* backB = WS((long)10*BN_*D_);
  float* difres= WS((long)10*BN_*D_);
  float* XsT   = WS((long)10*BN_*D_);
  float* Hb    = WS((long)10*BN_*D_);
  float* RH    = WS((long)13*BN_*D_);
  float* giB   = WS((long)BN_*3*D_);
  float* ghB   = WS((long)BN_*3*D_);
  float* qpB   = WS((long)13*BN_*D_);
  float* kpB   = WS((long)13*BN_*D_);
  float* vpB   = WS((long)13*BN_*D_);
  float* oAtt  = WS((long)10*BN_*D_);
  float* Zb    = WS((long)10*BN_*D_);
  float* predsI= WS((long)STEPS_*BN_*D_);
  float* gdec  = WS((long)BN_*D_);
  float* hzero = WS((long)BN_*D_);
  float* o1    = hs;   // reuse: layer scratch is dead by output stage
  float* o2    = Xw;

  EW(k_fill, (long)B_*STEPS_*N_*FK_, fk_sum, (long)B_*STEPS_*N_*FK_, 0.f);
  EW(k_fill, (long)BN_*D_, hzero, (long)BN_*D_, 0.f);

  // ---- embeddings / PE ----
  EW(k_time_embed, (long)B_*L_*N_, hd, TiD, DiW, TDb, DWb, hist);
  EW(k_pe, (long)64*D_, pe);

  // ---- dynamic graph learner ----
  EW(k_xs2, (long)BN_*L_, hist, Xs2);
  gemm(stream, Xs2, fc1W, fc1b, h64, BN_, 64, L_, L_, 64, 64, 0,0,0, 1,1,0, 1, 1.f);
  k_batchnorm<<<dim3(64), dim3(256), 0, stream>>>(h64, bn_g, bn_b, BN_, 64);
  gemm(stream, h64, fc2W, fc2b, dy, BN_, D_, 64, 64, D_, D_, 0,0,0, 1,1,0, 0, 1.f);

  // ---- static graph ----
  gemm(stream, Eu, Ed, nullptr, sta0, N_, N_, ND_, ND_, ND_, N_, 0,0,0, 1,1,1, 1, 1.f);
  k_softmax<<<dim3(N_), dim3(256), 0, stream>>>(sta0, N_);
  gemm(stream, sta0, sta0, nullptr, sta2, N_, N_, N_, N_, N_, N_, 0,0,0, 1,1,0, 0, 1.f);
  EW(k_mask, (long)N_*N_, sta0, stam0, 1);
  EW(k_mask, (long)N_*N_, sta2, stam1, 1);

  // ---- dynamic graphs (i=0 uses Ed, i=1 uses Eu) ----
  for (int i = 0; i < 2; ++i) {
    EW(k_build_xi, (long)BN_, dy, TDb, DWb, (i==0 ? Ed : Eu), Xi);
    gemm(stream, Xi, dWQ, nullptr, Qm, BN_, D_, AF_, AF_, D_, D_, 0,0,0, 1,1,0, 0, 1.f);
    gemm(stream, Xi, dWK, nullptr, Km, BN_, D_, AF_, AF_, D_, D_, 0,0,0, 1,1,0, 0, 1.f);
    gemm(stream, Qm, Km, nullptr, Alog, N_, N_, D_, D_, D_, N_,
         (long)N_*D_, (long)N_*D_, (long)N_*N_, B_, 1, 1, 0, 0.1767766953f);
    k_softmax<<<dim3(BN_), dim3(256), 0, stream>>>(Alog, N_);
    k_graph_norm<<<dim3(BN_), dim3(256), 0, stream>>>(Alog, adjs + (long)i*N_*N_,
                                                      gfull, gm + (long)(2*i)*BNN_);
    gemm(stream, gfull, gfull, nullptr, gpow, N_, N_, N_, N_, N_, N_,
         (long)N_*N_, (long)N_*N_, (long)N_*N_, B_, 1, 0, 0, 1.f);
    EW(k_mask, BNN_, gpow, gm + (long)(2*i+1)*BNN_, B_);
  }

  // ---- input embedding + gate features (layer-independent) ----
  EW(k_embed_cur, (long)B_*L_*N_*D_, hist, embW, embB, cur);
  EW(k_gate_feat, (long)B_*L_*N_, TDb, DWb, Eu, Ed, gFeat);

  // ================= layers =================
  for (int l = 0; l < NL_; ++l) {
    int LB = 15 + 27*l;
    const float* dif_backW = P(LB+0);  const float* dif_backb = P(LB+1);
    const float* dif_fkW   = P(LB+2);  const float* dif_fkb   = P(LB+3);
    const float* dif_lnb   = P(LB+4);  const float* dif_lng   = P(LB+5);
    const float* stc_fcW   = P(LB+6);  const float* stc_gcnW  = P(LB+7);
    const float* stc_gcnb  = P(LB+8);
    const float* gt_fc1W   = P(LB+9);  const float* gt_fc1b   = P(LB+10);
    const float* gt_fc2W   = P(LB+11); const float* gt_fc2b   = P(LB+12);
    const float* at_inW    = P(LB+13); const float* at_inb    = P(LB+14);
    const float* at_outW   = P(LB+15); const float* at_outb   = P(LB+16);
    const float* in_backW  = P(LB+17); const float* in_backb  = P(LB+18);
    const float* in_fkW    = P(LB+19); const float* in_fkb    = P(LB+20);
    const float* gr_Whh    = P(LB+21); const float* gr_Wih    = P(LB+22);
    const float* gr_bhh    = P(LB+23); const float* gr_bih    = P(LB+24);
    const float* in_lnb    = P(LB+25); const float* in_lng    = P(LB+26);

    int Lc = L_ - 2*l, L2 = Lc - 2;

    // --- gate ---
    gemm(stream, gFeat, gt_fc1W, gt_fc1b, gH, B_*L_*N_, 64, 40, 40, 64, 64, 0,0,0, 1,1,0, 1, 1.f);
    gemm(stream, gH, gt_fc2W, gt_fc2b, gO, B_*L_*N_, 1, 64, 64, 1, 1, 0,0,0, 1,1,0, 2, 1.f);
    EW(k_gate_apply, (long)B_*Lc*N_*D_, cur, Lc, gO, gated);

    // --- dif: main st_conv + decoder rollout ---
    st_conv(stream, gated, Lc, gm, stam0, stam1, stc_fcW, stc_gcnW, stc_gcnb,
            Xw, o96, Xsum, hs, hdif, (long)N_*D_);
    EW(k_slice, (long)B_*N_*D_, predsD, STEPS_, 0, hdif, L2, L2-1, 1);
    for (int s = 1; s < STEPS_; ++s) {
      int fromHist = KT_ - s; if (fromHist < 0) fromHist = 0;
      int fromPred = KT_ - fromHist;
      if (fromHist > 0)
        EW(k_slice, (long)B_*fromHist*N_*D_, Xcat, KT_, 0, gated, Lc, Lc-fromHist, fromHist);
      EW(k_slice, (long)B_*fromPred*N_*D_, Xcat, KT_, fromHist, predsD, STEPS_, 0, fromPred);
      st_conv(stream, Xcat, KT_, gm, stam0, stam1, stc_fcW, stc_gcnW, stc_gcnb,
              Xw, o96, Xsum, hs, predsD + (long)s*N_*D_, (long)STEPS_*N_*D_);
    }
    gemm(stream, predsD, dif_fkW, dif_fkb, fk_tmp, B_*STEPS_*N_, FK_, D_,
         D_, FK_, FK_, 0,0,0, 1,1,0, 0, 1.f);
    EW(k_add, (long)B_*STEPS_*N_*FK_, fk_sum, fk_tmp, (long)B_*STEPS_*N_*FK_);
    gemm(stream, hdif, dif_backW, dif_backb, backB, B_*L2*N_, D_, D_,
         D_, D_, D_, 0,0,0, 1,1,0, 0, 1.f);
    {
      long rows = (long)B_*L2*N_;
      k_sub_relu_ln<<<dim3((unsigned)cdiv(rows,8)), dim3(32,8), 0, stream>>>(
          cur, Lc, Lc-L2, backB, 0, L2, dif_lng, dif_lnb, difres);
    }

    // --- inh: GRU encode ---
    int Ll = L2;
    EW(k_btnd_lbnd, (long)Ll*BN_*D_, difres, XsT, Ll);
    for (int t = 0; t < Ll; ++t) {
      const float* hp = (t == 0) ? hzero : (Hb + (long)(t-1)*BN_*D_);
      gemm(stream, XsT + (long)t*BN_*D_, gr_Wih, gr_bih, giB, BN_, 3*D_, D_,
           D_, 3*D_, 3*D_, 0,0,0, 1,1,0, 0, 1.f);
      gemm(stream, hp, gr_Whh, gr_bhh, ghB, BN_, 3*D_, D_,
           D_, 3*D_, 3*D_, 0,0,0, 1,1,0, 0, 1.f);
      EW(k_gru, (long)BN_*D_, giB, ghB, hp, Hb + (long)t*BN_*D_);
    }
    EW(k_add_pe, (long)Ll*BN_*D_, Hb, pe, Ll);
    // self-attention over Hp
    gemm(stream, Hb, at_inW,      at_inb,      qpB, Ll*BN_, D_, D_, D_, 3*D_, D_, 0,0,0, 1,1,0, 0, 1.f);
    gemm(stream, Hb, at_inW+D_,   at_inb+D_,   kpB, Ll*BN_, D_, D_, D_, 3*D_, D_, 0,0,0, 1,1,0, 0, 1.f);
    gemm(stream, Hb, at_inW+2*D_, at_inb+2*D_, vpB, Ll*BN_, D_, D_, D_, 3*D_, D_, 0,0,0, 1,1,0, 0, 1.f);
    EW(k_attn, (long)BN_*HEADS_*Ll, qpB, kpB, vpB, oAtt, Ll, Ll);
    gemm(stream, oAtt, at_outW, at_outb, Zb, Ll*BN_, D_, D_, D_, D_, D_, 0,0,0, 1,1,0, 0, 1.f);
    EW(k_copy, (long)BN_*D_, predsI, Zb + (long)(Ll-1)*BN_*D_, (long)BN_*D_);
    EW(k_copy, (long)Ll*BN_*D_, RH, Hb, (long)Ll*BN_*D_);
    // decoder rollout
    for (int s = 1; s < STEPS_; ++s) {
      int curLen = Ll + s - 1;
      gemm(stream, predsI + (long)(s-1)*BN_*D_, gr_Wih, gr_bih, giB, BN_, 3*D_, D_,
           D_, 3*D_, 3*D_, 0,0,0, 1,1,0, 0, 1.f);
      gemm(stream, RH + (long)(curLen-1)*BN_*D_, gr_Whh, gr_bhh, ghB, BN_, 3*D_, D_,
           D_, 3*D_, 3*D_, 0,0,0, 1,1,0, 0, 1.f);
      EW(k_gru, (long)BN_*D_, giB, ghB, RH + (long)(curLen-1)*BN_*D_, gdec);
      EW(k_copy, (long)BN_*D_, RH + (long)curLen*BN_*D_, gdec, (long)BN_*D_);
      int newLen = curLen + 1;
      EW(k_add_pe, (long)newLen*BN_*D_, RH, pe, newLen);   // ref re-adds PE each step
      gemm(stream, gdec, at_inW,      at_inb,      qpB, BN_, D_, D_, D_, 3*D_, D_, 0,0,0, 1,1,0, 0, 1.f);
      gemm(stream, RH,   at_inW+D_,   at_inb+D_,   kpB, newLen*BN_, D_, D_, D_, 3*D_, D_, 0,0,0, 1,1,0, 0, 1.f);
      gemm(stream, RH,   at_inW+2*D_, at_inb+2*D_, vpB, newLen*BN_, D_, D_, D_, 3*D_, D_, 0,0,0, 1,1,0, 0, 1.f);
      EW(k_attn, (long)BN_*HEADS_, qpB, kpB, vpB, oAtt, 1, newLen);
      gemm(stream, oAtt, at_outW, at_outb, predsI + (long)s*BN_*D_, BN_, D_, D_,
           D_, D_, D_, 0,0,0, 1,1,0, 0, 1.f);
    }
    gemm(stream, predsI, in_fkW, in_fkb, fk_tmp, STEPS_*BN_, FK_, D_,
         D_, FK_, FK_, 0,0,0, 1,1,0, 0, 1.f);
    EW(k_fk_perm, (long)STEPS_*BN_*FK_, fk_sum, fk_tmp);
    gemm(stream, Zb, in_backW, in_backb, backB, Ll*BN_, D_, D_,
         D_, D_, D_, 0,0,0, 1,1,0, 0, 1.f);
    {
      long rows = (long)B_*Ll*N_;
      k_sub_relu_ln<<<dim3((unsigned)cdiv(rows,8)), dim3(32,8), 0, stream>>>(
          difres, Ll, 0, backB, 1, Ll, in_lng, in_lnb, cur);
    }
    // next layer: cur has length L2
  }

  // ---- output head ----
  EW(k_relu_copy, (long)B_*STEPS_*N_*FK_, fk_tmp, fk_sum, (long)B_*STEPS_*N_*FK_);
  gemm(stream, fk_tmp, out1W, out1b, o1, B_*STEPS_*N_, OUTH_, FK_,
       FK_, OUTH_, OUTH_, 0,0,0, 1,1,0, 1, 1.f);
  gemm(stream, o1, out2W, out2b, o2, B_*STEPS_*N_, GAP_, OUTH_,
       OUTH_, GAP_, GAP_, 0,0,0, 1,1,0, 0, 1.f);
  EW(k_final, (long)B_*STEPS_*N_*GAP_, o2, (float*)d_out);
}